// GIN_28741921144979
// MI455X (gfx1250) — compile-verified
//
#include <hip/hip_runtime.h>

typedef __attribute__((ext_vector_type(2))) float v2f;
typedef __attribute__((ext_vector_type(8))) float v8f;

#define HID 64
#define LDS_STRIDE 68   // 64 + 4 pad: kills bank conflicts for A-frag reads

// ---------------------------------------------------------------------------
// Tiny transpose: wt[n*K + k] = w[k*Nc + n]. Run once per weight per call so
// B-fragments become contiguous 8-byte loads (global_load_b64).
// ---------------------------------------------------------------------------
__global__ void gin_transpose_w(const float* __restrict__ w,
                                float* __restrict__ wt, int K, int Nc) {
    int t = blockIdx.x * blockDim.x + threadIdx.x;
    if (t >= K * Nc) return;
    int k = t / Nc;
    int n = t - k * Nc;
    wt[n * K + k] = w[k * Nc + n];
}

// ---------------------------------------------------------------------------
// Scatter-add: agg[dst] += x[src] over all edges. 16 lanes per edge, each
// lane gathers one float4 (contiguous 256B row -> coalesced) and does 4
// global f32 atomics into the L2-resident agg buffer.
// ---------------------------------------------------------------------------
__global__ void gin_scatter_add(const float* __restrict__ x,
                                const long long* __restrict__ ei,
                                float* __restrict__ agg, int nEdges) {
    int t = blockIdx.x * blockDim.x + threadIdx.x;
    int e = t >> 4;
    if (e >= nEdges) return;
    int c4 = t & 15;
    long long s = ei[e];
    long long d = ei[(long long)nEdges + e];
    float4 v = ((const float4*)(x + s * HID))[c4];
    float* a = agg + d * HID + c4 * 4;
    atomicAdd(a + 0, v.x);
    atomicAdd(a + 1, v.y);
    atomicAdd(a + 2, v.z);
    atomicAdd(a + 3, v.w);
}

// ---------------------------------------------------------------------------
// Fused GIN MLP for one 16-row tile per wave (32 threads / block):
//   hpre = (1+eps)*x + agg            (staged in LDS)
//   h1   = relu(hpre @ w1 + b1)       (WMMA f32 16x16x4, staged in LDS)
//   h2   = relu(h1   @ w2 + b2)       (WMMA f32 16x16x4)
//   accumulate BN sum / sumsq per column, write h2 tile coalesced.
// Weights come in TRANSPOSED (wt[n][k], stride 64) so each B-frag is one b64.
// ---------------------------------------------------------------------------
__global__ __launch_bounds__(32) void gin_mlp_wmma(
    const float* __restrict__ xin, const float* __restrict__ agg,
    const float* __restrict__ epsP,
    const float* __restrict__ wt1, const float* __restrict__ b1,
    const float* __restrict__ wt2, const float* __restrict__ b2,
    float* __restrict__ hout, float* __restrict__ stats, int nRows) {
    __shared__ float ldsIn[16 * LDS_STRIDE];
    __shared__ float ldsMid[16 * LDS_STRIDE];

    const int lane = threadIdx.x;          // 0..31
    const int rowBase = blockIdx.x * 16;
    if (rowBase >= nRows) return;          // uniform; grid divides exactly anyway

    const float epn = 1.0f + epsP[0];

    // Stage hpre tile (16x64) into LDS, coalesced float4 loads.
    #pragma unroll
    for (int i = 0; i < 8; ++i) {
        int linear = i * 32 + lane;        // 0..255 float4 slots
        int r  = linear >> 4;
        int c4 = linear & 15;
        long long row = (long long)(rowBase + r);
        float4 xv = ((const float4*)(xin + row * HID))[c4];
        float4 av = ((const float4*)(agg + row * HID))[c4];
        float4 h;
        h.x = epn * xv.x + av.x;
        h.y = epn * xv.y + av.y;
        h.z = epn * xv.z + av.z;
        h.w = epn * xv.w + av.w;
        *(float4*)&ldsIn[r * LDS_STRIDE + c4 * 4] = h;
    }
    __syncthreads();

    const int m  = lane & 15;              // A-frag row / B,C,D column-in-tile
    const int kh = lane >> 4;              // K-half select (upper lanes: K+2)

    // -------- GEMM1: ldsIn @ w1 + b1, relu -> ldsMid --------
    #pragma unroll
    for (int nn = 0; nn < 4; ++nn) {
        const int n0 = nn * 16;
        v8f acc;
        float bv = b1[n0 + m];
        #pragma unroll
        for (int r = 0; r < 8; ++r) acc[r] = bv;
        const float* wrow = wt1 + (n0 + m) * HID;   // transposed weight row
        #pragma unroll
        for (int k0 = 0; k0 < HID; k0 += 4) {
            v2f a, b;
            a.x = ldsIn[m * LDS_STRIDE + k0 + 2 * kh];
            a.y = ldsIn[m * LDS_STRIDE + k0 + 2 * kh + 1];
            b = *(const v2f*)(wrow + k0 + 2 * kh);  // one global_load_b64
            acc = __builtin_amdgcn_wmma_f32_16x16x4_f32(
                false, a, false, b, (short)0, acc, false, false);
        }
        #pragma unroll
        for (int r = 0; r < 8; ++r) {
            float v = acc[r] > 0.0f ? acc[r] : 0.0f;
            ldsMid[(r + 8 * kh) * LDS_STRIDE + n0 + m] = v;
        }
    }
    __syncthreads();

    // -------- GEMM2: ldsMid @ w2 + b2, relu -> ldsIn, BN stats --------
    #pragma unroll
    for (int nn = 0; nn < 4; ++nn) {
        const int n0 = nn * 16;
        v8f acc;
        float bv = b2[n0 + m];
        #pragma unroll
        for (int r = 0; r < 8; ++r) acc[r] = bv;
        const float* wrow = wt2 + (n0 + m) * HID;
        #pragma unroll
        for (int k0 = 0; k0 < HID; k0 += 4) {
            v2f a, b;
            a.x = ldsMid[m * LDS_STRIDE + k0 + 2 * kh];
            a.y = ldsMid[m * LDS_STRIDE + k0 + 2 * kh + 1];
            b = *(const v2f*)(wrow + k0 + 2 * kh);
            acc = __builtin_amdgcn_wmma_f32_16x16x4_f32(
                false, a, false, b, (short)0, acc, false, false);
        }
        float s = 0.0f, q = 0.0f;
        #pragma unroll
        for (int r = 0; r < 8; ++r) {
            float v = acc[r] > 0.0f ? acc[r] : 0.0f;
            s += v;
            q += v * v;
            ldsIn[(r + 8 * kh) * LDS_STRIDE + n0 + m] = v;
        }
        // lanes l and l+16 hold the same output column -> combine halves
        s += __shfl_xor(s, 16, 32);
        q += __shfl_xor(q, 16, 32);
        if (lane < 16) {
            atomicAdd(&stats[n0 + m], s);
            atomicAdd(&stats[HID + n0 + m], q);
        }
    }
    __syncthreads();

    // Coalesced float4 writeback of h2 tile.
    #pragma unroll
    for (int i = 0; i < 8; ++i) {
        int linear = i * 32 + lane;
        int r  = linear >> 4;
        int c4 = linear & 15;
        float4 v = *(const float4*)&ldsIn[r * LDS_STRIDE + c4 * 4];
        ((float4*)(hout + (long long)(rowBase + r) * HID))[c4] = v;
    }
}

// ---------------------------------------------------------------------------
// BatchNorm finalize: stats -> per-column scale/shift.
// ---------------------------------------------------------------------------
__global__ void gin_bn_finalize(const float* __restrict__ stats,
                                const float* __restrict__ g,
                                const float* __restrict__ bt,
                                float* __restrict__ ss, float invN) {
    int c = threadIdx.x;                   // 64 threads
    float mu  = stats[c] * invN;
    float var = stats[HID + c] * invN - mu * mu;
    float sc  = g[c] * rsqrtf(var + 1e-5f);
    ss[c]       = sc;
    ss[HID + c] = bt[c] - mu * sc;
}

// ---------------------------------------------------------------------------
// In-place affine + outer ReLU:  h = relu(h*scale + shift)
// ---------------------------------------------------------------------------
__global__ void gin_bn_apply_relu(float* __restrict__ h,
                                  const float* __restrict__ ss,
                                  long long totalF4) {
    long long t = (long long)blockIdx.x * blockDim.x + threadIdx.x;
    if (t >= totalF4) return;
    int c = (int)((t * 4) & (HID - 1));    // column of first element
    float4 v = ((float4*)h)[t];
    v.x = v.x * ss[c + 0] + ss[HID + c + 0];
    v.y = v.y * ss[c + 1] + ss[HID + c + 1];
    v.z = v.z * ss[c + 2] + ss[HID + c + 2];
    v.w = v.w * ss[c + 3] + ss[HID + c + 3];
    v.x = v.x > 0.0f ? v.x : 0.0f;
    v.y = v.y > 0.0f ? v.y : 0.0f;
    v.z = v.z > 0.0f ? v.z : 0.0f;
    v.w = v.w > 0.0f ? v.w : 0.0f;
    ((float4*)h)[t] = v;
}

// ---------------------------------------------------------------------------
// Final linear: out(16 cols) = x @ wl + bl via WMMA f32 16x16x4.
// wlt is transposed (wlt[n*64 + k]).
// ---------------------------------------------------------------------------
__global__ __launch_bounds__(32) void gin_final_linear(
    const float* __restrict__ x, const float* __restrict__ wlt,
    const float* __restrict__ bl, float* __restrict__ out, int nRows) {
    const int lane = threadIdx.x;
    const int rowBase = blockIdx.x * 16;
    if (rowBase >= nRows) return;
    const int m  = lane & 15;
    const int kh = lane >> 4;
    v8f acc;
    float bv = bl[m];
    #pragma unroll
    for (int r = 0; r < 8; ++r) acc[r] = bv;
    const float* wrow = wlt + m * HID;
    #pragma unroll
    for (int k0 = 0; k0 < HID; k0 += 4) {
        v2f a, b;
        long long row = (long long)(rowBase + m);
        a.x = x[row * HID + k0 + 2 * kh];
        a.y = x[row * HID + k0 + 2 * kh + 1];
        b = *(const v2f*)(wrow + k0 + 2 * kh);
        acc = __builtin_amdgcn_wmma_f32_16x16x4_f32(
            false, a, false, b, (short)0, acc, false, false);
    }
    #pragma unroll
    for (int r = 0; r < 8; ++r)
        out[(long long)(rowBase + r + 8 * kh) * 16 + m] = acc[r];
}

// ---------------------------------------------------------------------------
extern "C" void kernel_launch(void* const* d_in, const int* in_sizes, int n_in,
                              void* d_out, int out_size, void* d_ws, size_t ws_size,
                              hipStream_t stream) {
    const int N = in_sizes[0] / HID;
    const int E = in_sizes[1] / 2;

    const float* x0 = (const float*)d_in[0];
    const long long* ei = (const long long*)d_in[1];
    auto P = [&](int i) { return (const float*)d_in[i]; };

    float* ws = (float*)d_ws;
    const long long NF = (long long)N * HID;
    float* agg   = ws;                       // N*64
    float* bufA  = ws + NF;                  // N*64
    float* bufB  = ws + 2 * NF;              // N*64
    float* stats = ws + 3 * NF;              // 128
    float* ss    = ws + 3 * NF + 128;        // 128
    float* wt1   = ws + 3 * NF + 256;        // 64*64 transposed
    float* wt2   = wt1 + HID * HID;          // 64*64 transposed
    float* wlt   = wt2 + HID * HID;          // 16*64 transposed

    const int scatterBlocks = (E * 16 + 255) / 256;
    const int tileBlocks    = (N + 15) / 16;
    const long long totalF4 = NF / 4;
    const int ewBlocks      = (int)((totalF4 + 255) / 256);
    const float invN = 1.0f / (float)N;

    const float* xcur = x0;
    float* hb[3] = {bufA, bufB, bufA};

    for (int layer = 0; layer < 3; ++layer) {
        const int base = 2 + 7 * layer;
        const float* eps = P(base + 0);
        const float* w1  = P(base + 1);
        const float* b1  = P(base + 2);
        const float* w2  = P(base + 3);
        const float* b2  = P(base + 4);
        const float* g   = P(base + 5);
        const float* bt  = P(base + 6);
        float* hbuf = hb[layer];

        hipMemsetAsync(agg, 0, NF * sizeof(float), stream);
        hipMemsetAsync(stats, 0, 128 * sizeof(float), stream);

        gin_transpose_w<<<16, 256, 0, stream>>>(w1, wt1, HID, HID);
        gin_transpose_w<<<16, 256, 0, stream>>>(w2, wt2, HID, HID);

        gin_scatter_add<<<scatterBlocks, 256, 0, stream>>>(xcur, ei, agg, E);
        gin_mlp_wmma<<<tileBlocks, 32, 0, stream>>>(xcur, agg, eps, wt1, b1,
                                                    wt2, b2, hbuf, stats, N);
        gin_bn_finalize<<<1, HID, 0, stream>>>(stats, g, bt, ss, invN);
        gin_bn_apply_relu<<<ewBlocks, 256, 0, stream>>>(hbuf, ss, totalF4);

        xcur = hbuf;
    }

    gin_transpose_w<<<4, 256, 0, stream>>>((const float*)d_in[23], wlt, HID, 16);
    gin_final_linear<<<tileBlocks, 32, 0, stream>>>(xcur, wlt,
                                                    (const float*)d_in[24],
                                                    (float*)d_out, N);
}